// GraphConstruction_30356828848494
// MI455X (gfx1250) — compile-verified
//
#include <hip/hip_runtime.h>
#include <hip/hip_bf16.h>
#include <math.h>

typedef __attribute__((ext_vector_type(2))) float v2f;
typedef __attribute__((ext_vector_type(8))) float v8f;

#define RADIUS_C 0.1f
#define FW 0.5f
#define BM 64
#define BN 64
#define BK 32

// ---------------- init: zero the two max accumulators ----------------
__global__ void init_ws(unsigned int* maxes) {
    if (threadIdx.x < 2) maxes[threadIdx.x] = 0u;
}

// ---------------- semantic row norms: one block per row ----------------
__global__ __launch_bounds__(256)
void row_norms(const float* __restrict__ sem, float* __restrict__ norms, int N, int D) {
    __shared__ float red[256];
    int row = blockIdx.x;
    const float* r = sem + (size_t)row * D;
    float s = 0.f;
    for (int c = threadIdx.x; c < D; c += 256) { float x = r[c]; s += x * x; }
    red[threadIdx.x] = s;
    __syncthreads();
    for (int off = 128; off > 0; off >>= 1) {
        if (threadIdx.x < off) red[threadIdx.x] += red[threadIdx.x + off];
        __syncthreads();
    }
    if (threadIdx.x == 0) norms[row] = red[0];
}

// ---------------- spatial max distance (D=2, input L2-resident) ----------------
__global__ __launch_bounds__(256)
void spatial_max_kernel(const float* __restrict__ sp, unsigned int* __restrict__ maxes, int N) {
    __shared__ float red[256];
    size_t total = (size_t)N * N;
    size_t stride = (size_t)gridDim.x * 256;
    float m = 0.f;
    for (size_t e = (size_t)blockIdx.x * 256 + threadIdx.x; e < total; e += stride) {
        int i = (int)(e / (size_t)N);
        int j = (int)(e % (size_t)N);
        float dx = sp[2 * i] - sp[2 * j];
        float dy = sp[2 * i + 1] - sp[2 * j + 1];
        float d2 = dx * dx + dy * dy;
        float d = d2 > 0.f ? sqrtf(fmaxf(d2, 1e-12f)) : 0.f;
        m = fmaxf(m, d);
    }
    red[threadIdx.x] = m;
    __syncthreads();
    for (int off = 128; off > 0; off >>= 1) {
        if (threadIdx.x < off) red[threadIdx.x] = fmaxf(red[threadIdx.x], red[threadIdx.x + off]);
        __syncthreads();
    }
    if (threadIdx.x == 0) atomicMax(maxes + 1, __float_as_uint(red[0]));
}

// ---------------- semantic distance via fp32 WMMA GEMM ----------------
// S = A * A^T, d = sqrt(max(ni + nj - 2*S, 0)); writes raw dist into rawOut,
// atomic-max of raw dist into maxes[0]. Block tile 64x64, K-tile 32, 8 waves,
// each wave owns two 16x16 accumulators (same N-tile, adjacent M-tiles).
__global__ __launch_bounds__(256)
void sem_dist_wmma(const float* __restrict__ A, const float* __restrict__ norms,
                   float* __restrict__ rawOut, unsigned int* __restrict__ maxes,
                   int N, int D) {
    __shared__ float As[BM * BK];
    __shared__ float Bs[BN * BK];

    const int tid  = threadIdx.x;
    const int lane = tid & 31;
    const int wave = tid >> 5;
    const int half = (lane >> 4) & 1;   // lanes 16-31 carry K+2 / M+8
    const int m16  = lane & 15;
    const int tc   = wave & 3;          // tile column 0..3
    const int tr0  = (wave >> 2) * 2;   // tile rows tr0, tr0+1

    const size_t rowBase = (size_t)blockIdx.y * BM;
    const size_t colBase = (size_t)blockIdx.x * BN;

    v8f acc0 = {};
    v8f acc1 = {};

    for (int k0 = 0; k0 < D; k0 += BK) {
        // Stage A tile (rows rowBase..+64) and B tile (rows colBase..+64, B = A^T
        // of the column block, stored [n][k]); 512 float4 per tile, 2 per thread.
#pragma unroll
        for (int i = 0; i < 2; ++i) {
            int l4 = i * 256 + tid;          // 0..511
            int r  = l4 >> 3;                // row within tile
            int c4 = (l4 & 7) << 2;          // float offset within K-tile
            *(float4*)(&As[r * BK + c4]) =
                *(const float4*)(A + (rowBase + r) * (size_t)D + k0 + c4);
            *(float4*)(&Bs[r * BK + c4]) =
                *(const float4*)(A + (colBase + r) * (size_t)D + k0 + c4);
        }
        __syncthreads();

#pragma unroll
        for (int kk = 0; kk < BK; kk += 4) {
            int kb = kk + 2 * half;
            // B-frag (4x16): lanes 0-15 hold K=kk,kk+1; lanes 16-31 hold K=kk+2,kk+3
            v2f b;
            b.x = Bs[(tc * 16 + m16) * BK + kb];
            b.y = Bs[(tc * 16 + m16) * BK + kb + 1];
            // A-frags (16x4), same K striping
            v2f a0, a1;
            a0.x = As[(tr0 * 16 + m16) * BK + kb];
            a0.y = As[(tr0 * 16 + m16) * BK + kb + 1];
            a1.x = As[((tr0 + 1) * 16 + m16) * BK + kb];
            a1.y = As[((tr0 + 1) * 16 + m16) * BK + kb + 1];

            acc0 = __builtin_amdgcn_wmma_f32_16x16x4_f32(
                false, a0, false, b, (short)0, acc0, false, false);
            acc1 = __builtin_amdgcn_wmma_f32_16x16x4_f32(
                false, a1, false, b, (short)0, acc1, false, false);
        }
        __syncthreads();
    }

    // Epilogue: Gram -> raw Euclidean distance; track local max.
    float lmax = 0.f;
    const size_t gj = colBase + tc * 16 + m16;
    const float nj = norms[gj];
#pragma unroll
    for (int v = 0; v < 8; ++v) {
        {
            size_t gi = rowBase + (size_t)(tr0 * 16 + v + 8 * half);
            float d2 = fmaxf(norms[gi] + nj - 2.f * acc0[v], 0.f);
            float d = d2 > 0.f ? sqrtf(fmaxf(d2, 1e-12f)) : 0.f;
            rawOut[gi * (size_t)N + gj] = d;
            lmax = fmaxf(lmax, d);
        }
        {
            size_t gi = rowBase + (size_t)((tr0 + 1) * 16 + v + 8 * half);
            float d2 = fmaxf(norms[gi] + nj - 2.f * acc1[v], 0.f);
            float d = d2 > 0.f ? sqrtf(fmaxf(d2, 1e-12f)) : 0.f;
            rawOut[gi * (size_t)N + gj] = d;
            lmax = fmaxf(lmax, d);
        }
    }

    __syncthreads();
    As[tid] = lmax;
    __syncthreads();
    for (int off = 128; off > 0; off >>= 1) {
        if (tid < off) As[tid] = fmaxf(As[tid], As[tid + off]);
        __syncthreads();
    }
    if (tid == 0) atomicMax(maxes, __float_as_uint(As[0]));
}

// ---------------- fuse: normalize, combine, radius adjacency ----------------
__global__ __launch_bounds__(256)
void combine_kernel(const float* __restrict__ sp,
                    const unsigned int* __restrict__ maxes,
                    float* __restrict__ comb, int* __restrict__ adj, int N) {
    int j = blockIdx.x * 256 + threadIdx.x;
    int i = blockIdx.y;
    float semMax = __uint_as_float(maxes[0]);
    float spMax  = __uint_as_float(maxes[1]);
    float dx = sp[2 * i] - sp[2 * j];
    float dy = sp[2 * i + 1] - sp[2 * j + 1];
    float d2 = dx * dx + dy * dy;
    float sd = d2 > 0.f ? sqrtf(fmaxf(d2, 1e-12f)) : 0.f;
    size_t idx = (size_t)i * N + j;
    float se = comb[idx];  // raw semantic distance staged in d_out
    float sdn = spMax  > 0.f ? sd / fmaxf(spMax,  1e-30f) : sd;
    float sen = semMax > 0.f ? se / fmaxf(semMax, 1e-30f) : se;
    float c = (1.f - FW) * sdn + FW * sen;
    comb[idx] = c;
    adj[idx] = (c < RADIUS_C && i != j) ? 1 : 0;
}

// ---------------- per-row top-9 (drop self) ----------------
__global__ __launch_bounds__(256)
void knn_kernel(const float* __restrict__ comb, int* __restrict__ knn, int N) {
    const int L = 9;
    __shared__ float sv[256 * L];
    __shared__ int   si[256 * L];
    __shared__ float rv[256];
    __shared__ int   ri[256];
    __shared__ int   rp[256];
    __shared__ int   sel[L];

    int row = blockIdx.x, tid = threadIdx.x;
    const float* r = comb + (size_t)row * N;

    float v[L];
    int ix[L];
#pragma unroll
    for (int k = 0; k < L; ++k) { v[k] = 3.4e38f; ix[k] = -1; }

    for (int j = tid; j < N; j += 256) {
        float d = r[j];
        if (d < v[L - 1]) {
            int p = L - 1;
            while (p > 0 && v[p - 1] > d) {
                v[p] = v[p - 1]; ix[p] = ix[p - 1]; --p;
            }
            v[p] = d; ix[p] = j;
        }
    }
#pragma unroll
    for (int k = 0; k < L; ++k) { sv[tid * L + k] = v[k]; si[tid * L + k] = ix[k]; }
    __syncthreads();

    for (int round = 0; round < L; ++round) {
        float bv = 3.4e38f; int bi = 0x7fffffff, bp = -1;
#pragma unroll
        for (int k = 0; k < L; ++k) {
            float x = sv[tid * L + k];
            int  xi = si[tid * L + k];
            if (x < bv || (x == bv && xi >= 0 && xi < bi)) { bv = x; bi = xi; bp = tid * L + k; }
        }
        rv[tid] = bv; ri[tid] = bi; rp[tid] = bp;
        __syncthreads();
        for (int off = 128; off > 0; off >>= 1) {
            if (tid < off) {
                float ov = rv[tid + off]; int oi = ri[tid + off];
                if (ov < rv[tid] || (ov == rv[tid] && oi < ri[tid])) {
                    rv[tid] = ov; ri[tid] = oi; rp[tid] = rp[tid + off];
                }
            }
            __syncthreads();
        }
        if (tid == 0) {
            sel[round] = ri[0];
            if (rp[0] >= 0) sv[rp[0]] = 3.4e38f;
        }
        __syncthreads();
    }
    if (tid < L - 1) knn[(size_t)row * (L - 1) + tid] = sel[tid + 1];  // drop nearest (self)
}

// ---------------- scatter kNN edges both directions ----------------
__global__ void knn_scatter(const int* __restrict__ knn, int* __restrict__ adj, int N) {
    int e = blockIdx.x * 256 + threadIdx.x;
    if (e >= N * 8) return;
    int i = e >> 3;
    int j = knn[e];
    if (j >= 0 && j < N && j != i) {
        adj[(size_t)i * N + j] = 1;
        adj[(size_t)j * N + i] = 1;
    }
}

extern "C" void kernel_launch(void* const* d_in, const int* in_sizes, int n_in,
                              void* d_out, int out_size, void* d_ws, size_t ws_size,
                              hipStream_t stream) {
    const float* sp  = (const float*)d_in[0];   // [N,2]
    const float* sem = (const float*)d_in[1];   // [N,512]
    int N = in_sizes[0] / 2;
    int D = in_sizes[1] / N;

    float* comb = (float*)d_out;                       // [N*N] combined (f32)
    int*   adj  = (int*)d_out + (size_t)N * N;         // [N*N] adjacency (i32)

    unsigned int* maxes = (unsigned int*)d_ws;         // [0]=semMax bits, [1]=spMax bits
    float* norms = (float*)d_ws + 4;                   // N floats
    int*   knn   = (int*)((float*)d_ws + 4 + N);       // N*8 ints

    init_ws<<<1, 64, 0, stream>>>(maxes);
    row_norms<<<N, 256, 0, stream>>>(sem, norms, N, D);
    spatial_max_kernel<<<2048, 256, 0, stream>>>(sp, maxes, N);

    dim3 gg(N / BN, N / BM);
    sem_dist_wmma<<<gg, 256, 0, stream>>>(sem, norms, comb, maxes, N, D);

    dim3 gc(N / 256, N);
    combine_kernel<<<gc, 256, 0, stream>>>(sp, maxes, comb, adj, N);

    knn_kernel<<<N, 256, 0, stream>>>(comb, knn, N);
    knn_scatter<<<(N * 8 + 255) / 256, 256, 0, stream>>>(knn, adj, N);
}